// KATRec_3676492005469
// MI455X (gfx1250) — compile-verified
//
#include <hip/hip_runtime.h>
#include <hip/hip_bf16.h>

#define NUM_ITEMS    50000
#define NUM_USERS    20000
#define NUM_ENT      100000
#define N_NODES      (NUM_USERS + NUM_ENT)
#define NNZ          1000000
#define DD           64
#define HH           2
#define HDIM         32
#define LL           50
#define BB           512
#define ALPHA_KG     0.2f
#define NEGV         (-1e9f)
#define EPSV         1e-12f
#define SCALE_ATT    0.17677669529663687f   /* 1/sqrt(32) */

typedef float v2f __attribute__((ext_vector_type(2)));
typedef float v8f __attribute__((ext_vector_type(8)));

// ---------- wave32 reductions ----------
__device__ __forceinline__ float wsum(float v) {
#pragma unroll
  for (int o = 16; o > 0; o >>= 1) v += __shfl_xor(v, o, 32);
  return v;
}
__device__ __forceinline__ float wmaxr(float v) {
#pragma unroll
  for (int o = 16; o > 0; o >>= 1) v = fmaxf(v, __shfl_xor(v, o, 32));
  return v;
}

// ---------- GCN: one wave per edge, float2 per lane, native f32 atomics ----------
__global__ __launch_bounds__(256) void k_spmv(
    const int* __restrict__ rows, const int* __restrict__ cols,
    const float* __restrict__ vals, const float* __restrict__ xu,
    const float* __restrict__ xe, int split, float* __restrict__ out, int nnz) {
  int e = blockIdx.x * (blockDim.x >> 5) + (threadIdx.x >> 5);
  if (e >= nnz) return;
  int lane = threadIdx.x & 31;
  int c = cols[e], r = rows[e];
  float v = vals[e];
  const float* src = (c < split) ? (xu + (long)c * DD) : (xe + (long)(c - split) * DD);
  float2 s = *(const float2*)(src + lane * 2);
  float* dst = out + (long)r * DD + lane * 2;
  __hip_atomic_fetch_add(dst,     s.x * v, __ATOMIC_RELAXED, __HIP_MEMORY_SCOPE_AGENT);
  __hip_atomic_fetch_add(dst + 1, s.y * v, __ATOMIC_RELAXED, __HIP_MEMORY_SCOPE_AGENT);
}

__global__ __launch_bounds__(256) void k_mean(
    const float* __restrict__ ent0, const float* __restrict__ x1,
    const float* __restrict__ x2, float* __restrict__ z) {
  long i = (long)blockIdx.x * blockDim.x + threadIdx.x;
  long n = (long)NUM_ENT * DD;
  if (i >= n) return;
  long j = (long)NUM_USERS * DD + i;
  z[i] = (ent0[i] + x1[j] + x2[j]) * (1.0f / 3.0f);
}

// ---------- gathers ----------
__global__ __launch_bounds__(256) void k_gather_tok(
    const int* __restrict__ seq, const int* __restrict__ i2e,
    const float* __restrict__ z, float* __restrict__ entk, int* __restrict__ valid) {
  int t = blockIdx.x * (blockDim.x >> 5) + (threadIdx.x >> 5);
  if (t >= BB * LL) return;
  int lane = threadIdx.x & 31;
  int s = seq[t];
  int idx0 = (s > 0) ? s - 1 : 0;
  int eid = i2e[idx0];
  int val = (eid >= 0) && (s > 0);
  if (lane == 0) valid[t] = val;
  float2 r = make_float2(0.f, 0.f);
  if (val) r = *(const float2*)(z + (long)eid * DD + lane * 2);
  *(float2*)(entk + (long)t * DD + lane * 2) = r;
}

__global__ __launch_bounds__(256) void k_gather_items(
    const int* __restrict__ i2e, const float* __restrict__ z, float* __restrict__ aek) {
  int t = blockIdx.x * (blockDim.x >> 5) + (threadIdx.x >> 5);
  if (t >= NUM_ITEMS) return;
  int lane = threadIdx.x & 31;
  int eid = i2e[t];
  float2 r = make_float2(0.f, 0.f);
  if (eid >= 0) r = *(const float2*)(z + (long)eid * DD + lane * 2);
  *(float2*)(aek + (long)t * DD + lane * 2) = r;
}

// ---------- WMMA f32 GEMM ----------
// One wave per block owns a 16(M) x 64(N) stripe: 4 independent accumulators,
// A tile loaded once per K-step and reused by 4 WMMAs. FLAGS is a template
// parameter (branch-free inner loop). All addressing uses 32-bit offsets off
// the kernarg base pointers -> global_load saddr form, no 64-bit VALU adds.
#define GF_RELU 1
#define GF_ACC  2
#define GF_TRB  4
#define GF_NT   8
template <int FLAGS>
__global__ __launch_bounds__(32) void k_gemm16(
    const float* __restrict__ A, const float* __restrict__ Bm,
    const float* __restrict__ bias, float* __restrict__ C,
    int N, int K, int lda, int ldb, int ldc) {
  int lane = threadIdx.x & 31;
  int ntiles = (N + 15) >> 4;
  int tm = blockIdx.y;
  int m  = lane & 15;                 // M row (A) / N col (B,C,D) for this lane
  int kh = (lane >> 4) * 2;           // K half select per ISA f32 A/B layout
  int aOff = (tm * 16 + m) * lda + kh;
  int t0 = blockIdx.x * 4;
  int tc0 = min(t0 + 0, ntiles - 1);  // clamp: dup compute, store guarded
  int tc1 = min(t0 + 1, ntiles - 1);
  int tc2 = min(t0 + 2, ntiles - 1);
  int tc3 = min(t0 + 3, ntiles - 1);
  int bo0, bo1, bo2, bo3, bstep;
  if (FLAGS & GF_TRB) {               // B stored [N,K] row-major (B^T logical)
    bo0 = (tc0 * 16 + m) * ldb + kh;
    bo1 = (tc1 * 16 + m) * ldb + kh;
    bo2 = (tc2 * 16 + m) * ldb + kh;
    bo3 = (tc3 * 16 + m) * ldb + kh;
    bstep = 4;
  } else {                            // B stored [K,N] row-major
    bo0 = kh * ldb + tc0 * 16 + m;
    bo1 = kh * ldb + tc1 * 16 + m;
    bo2 = kh * ldb + tc2 * 16 + m;
    bo3 = kh * ldb + tc3 * 16 + m;
    bstep = 4 * ldb;
  }
  v8f c0 = {0.f, 0.f, 0.f, 0.f, 0.f, 0.f, 0.f, 0.f};
  v8f c1 = c0, c2 = c0, c3 = c0;
  for (int k0 = 0; k0 < K; k0 += 4) {
    v2f a = *(const v2f*)(A + aOff);
    aOff += 4;
    v2f b0, b1, b2, b3;
    if (FLAGS & GF_TRB) {
      b0 = *(const v2f*)(Bm + bo0);
      b1 = *(const v2f*)(Bm + bo1);
      b2 = *(const v2f*)(Bm + bo2);
      b3 = *(const v2f*)(Bm + bo3);
    } else {
      b0.x = Bm[bo0]; b0.y = Bm[bo0 + ldb];
      b1.x = Bm[bo1]; b1.y = Bm[bo1 + ldb];
      b2.x = Bm[bo2]; b2.y = Bm[bo2 + ldb];
      b3.x = Bm[bo3]; b3.y = Bm[bo3 + ldb];
    }
    bo0 += bstep; bo1 += bstep; bo2 += bstep; bo3 += bstep;
    c0 = __builtin_amdgcn_wmma_f32_16x16x4_f32(false, a, false, b0, (short)0, c0, false, false);
    c1 = __builtin_amdgcn_wmma_f32_16x16x4_f32(false, a, false, b1, (short)0, c1, false, false);
    c2 = __builtin_amdgcn_wmma_f32_16x16x4_f32(false, a, false, b2, (short)0, c2, false, false);
    c3 = __builtin_amdgcn_wmma_f32_16x16x4_f32(false, a, false, b3, (short)0, c3, false, false);
  }
  int mrow = tm * 16 + ((lane >> 4) * 8);  // C/D layout: VGPR v -> row v (+8 hi)
#define EPI(J, CJ)                                                        \
  if (t0 + (J) < ntiles) {                                                \
    int n = (t0 + (J)) * 16 + m;                                          \
    float bv = bias ? bias[n] : 0.f;                                      \
    _Pragma("unroll") for (int v = 0; v < 8; v++) {                       \
      int off = (mrow + v) * ldc + n;                                     \
      float val = CJ[v] + bv;                                             \
      if (FLAGS & GF_ACC) val += C[off];                                  \
      if (FLAGS & GF_RELU) val = fmaxf(val, 0.f);                         \
      if (FLAGS & GF_NT) __builtin_nontemporal_store(val, C + off);       \
      else C[off] = val;                                                  \
    }                                                                     \
  }
  EPI(0, c0) EPI(1, c1) EPI(2, c2) EPI(3, c3)
#undef EPI
}

// ---------- KG attention bias + causal mask ----------
__global__ __launch_bounds__(256) void k_mask(
    const float* __restrict__ qb, const float* __restrict__ kb,
    const int* __restrict__ valid, float* __restrict__ mask) {
  long i = (long)blockIdx.x * blockDim.x + threadIdx.x;
  long n = (long)BB * HH * LL * LL;
  if (i >= n) return;
  int k = i % LL; long r = i / LL;
  int q = r % LL; r /= LL;
  int hh = r % HH; int b = (int)(r / HH);
  float bias;
  if (!valid[b * LL + q] || !valid[b * LL + k]) {
    bias = NEGV;
  } else {
    const float* qr = qb + (long)(b * LL + q) * DD + hh * HDIM;
    const float* kr = kb + (long)(b * LL + k) * DD + hh * HDIM;
    float s = 0.f;
#pragma unroll
    for (int d = 0; d < HDIM; d++) s += qr[d] * kr[d];
    bias = s * SCALE_ATT;
  }
  float causal = (k <= q) ? 0.f : NEGV;
  mask[i] = causal + ALPHA_KG * bias;
}

// ---------- embedding + initial layernorm (wave per token, LDS-free) ----------
__global__ __launch_bounds__(256) void k_hinit(
    const int* __restrict__ seq, const float* __restrict__ item_emb,
    const float* __restrict__ pos_emb, const float* __restrict__ g,
    const float* __restrict__ bb, float* __restrict__ h) {
  int t = blockIdx.x * (blockDim.x >> 5) + (threadIdx.x >> 5);
  if (t >= BB * LL) return;
  int lane = threadIdx.x & 31;
  int l = t % LL;
  int s = seq[t];
  int idx0 = (s > 0) ? s - 1 : 0;
  float np = (s > 0) ? 1.f : 0.f;
  float2 ie = *(const float2*)(item_emb + (long)idx0 * DD + lane * 2);
  float2 pe = *(const float2*)(pos_emb + (long)l * DD + lane * 2);
  float x0 = ie.x * np + pe.x, x1 = ie.y * np + pe.y;
  float mean = wsum(x0 + x1) * (1.f / 64.f);
  float d0 = x0 - mean, d1 = x1 - mean;
  float var = wsum(d0 * d0 + d1 * d1) * (1.f / 64.f);
  float inv = rsqrtf(var + EPSV);
  int c = lane * 2;
  h[(long)t * DD + c]     = d0 * inv * g[c] + bb[c];
  h[(long)t * DD + c + 1] = d1 * inv * g[c + 1] + bb[c + 1];
}

// ---------- residual add + layernorm (+ optional pad zeroing) ----------
__global__ __launch_bounds__(256) void k_addln(
    float* __restrict__ h, const float* __restrict__ delta,
    const float* __restrict__ g, const float* __restrict__ bb,
    const int* __restrict__ seq, int zero_pads) {
  int t = blockIdx.x * (blockDim.x >> 5) + (threadIdx.x >> 5);
  if (t >= BB * LL) return;
  int lane = threadIdx.x & 31;
  float2 hv = *(const float2*)(h + (long)t * DD + lane * 2);
  float2 dv = *(const float2*)(delta + (long)t * DD + lane * 2);
  float x0 = hv.x + dv.x, x1 = hv.y + dv.y;
  float mean = wsum(x0 + x1) * (1.f / 64.f);
  float d0 = x0 - mean, d1 = x1 - mean;
  float var = wsum(d0 * d0 + d1 * d1) * (1.f / 64.f);
  float inv = rsqrtf(var + EPSV);
  int c = lane * 2;
  float y0 = d0 * inv * g[c] + bb[c];
  float y1 = d1 * inv * g[c + 1] + bb[c + 1];
  if (zero_pads && seq[t] == 0) { y0 = 0.f; y1 = 0.f; }
  h[(long)t * DD + c]     = y0;
  h[(long)t * DD + c + 1] = y1;
}

// ---------- fused scores + softmax: one wave per (b,h,q) row ----------
__global__ __launch_bounds__(128) void k_attn(
    const float* __restrict__ Q, const float* __restrict__ Kb,
    const float* __restrict__ mask, const int* __restrict__ seq,
    float* __restrict__ att) {
  int w = blockIdx.x * (blockDim.x >> 5) + (threadIdx.x >> 5);
  if (w >= BB * HH * LL) return;
  int lane = threadIdx.x & 31;
  int q = w % LL; int r = w / LL; int hh = r % HH; int b = r / HH;
  const float* qr = Q + (long)(b * LL + q) * DD + hh * HDIM;
  long mbase = (long)w * LL;
  int k0 = lane, k1 = lane + 32;
  float sc0, sc1 = -3.0e38f;
  {
    const float* kr = Kb + (long)(b * LL + k0) * DD + hh * HDIM;
    float s = 0.f;
#pragma unroll
    for (int d = 0; d < HDIM; d++) s += qr[d] * kr[d];
    s = s * SCALE_ATT + mask[mbase + k0];
    if (seq[b * LL + k0] == 0) s = NEGV;
    sc0 = s;
  }
  if (k1 < LL) {
    const float* kr = Kb + (long)(b * LL + k1) * DD + hh * HDIM;
    float s = 0.f;
#pragma unroll
    for (int d = 0; d < HDIM; d++) s += qr[d] * kr[d];
    s = s * SCALE_ATT + mask[mbase + k1];
    if (seq[b * LL + k1] == 0) s = NEGV;
    sc1 = s;
  }
  float mx = wmaxr(fmaxf(sc0, sc1));
  float e0 = __expf(sc0 - mx);
  float e1 = (k1 < LL) ? __expf(sc1 - mx) : 0.f;
  float inv = 1.f / wsum(e0 + e1);
  att[mbase + k0] = e0 * inv;
  if (k1 < LL) att[mbase + k1] = e1 * inv;
}

// ---------- ctx = attn @ V : one thread per output element ----------
__global__ __launch_bounds__(256) void k_ctx(
    const float* __restrict__ att, const float* __restrict__ V,
    float* __restrict__ ctx) {
  long i = (long)blockIdx.x * blockDim.x + threadIdx.x;
  long n = (long)BB * LL * DD;
  if (i >= n) return;
  int d = i % HDIM; long r = i / HDIM;
  int hh = r % HH; r /= HH;
  int q = r % LL; int b = (int)(r / LL);
  const float* ar = att + ((long)(b * HH + hh) * LL + q) * LL;
  const float* vr = V + (long)b * LL * DD + hh * HDIM + d;
  float s = 0.f;
#pragma unroll
  for (int k = 0; k < LL; k++) s += ar[k] * vr[(long)k * DD];
  ctx[(long)(b * LL + q) * DD + hh * HDIM + d] = s;
}

// ---------- last valid position -> user_vec ----------
__global__ __launch_bounds__(256) void k_lastpos(
    const int* __restrict__ seq, const float* __restrict__ h, float* __restrict__ uv) {
  int b = blockIdx.x * (blockDim.x >> 5) + (threadIdx.x >> 5);
  if (b >= BB) return;
  int lane = threadIdx.x & 31;
  int c0 = (lane < LL) ? (seq[b * LL + lane] > 0) : 0;
  int c1 = (lane + 32 < LL) ? (seq[b * LL + lane + 32] > 0) : 0;
  int len = (int)wsum((float)(c0 + c1));
  if (len < 1) len = 1;
  const float* src = h + ((long)b * LL + (len - 1)) * DD;
  uv[(long)b * DD + lane]      = src[lane];
  uv[(long)b * DD + lane + 32] = src[lane + 32];
}

// ---------- host orchestration ----------
static inline void gemm(hipStream_t s, const float* A, const float* B,
                        const float* bias, float* C, int M, int N, int K,
                        int lda, int ldb, int ldc, int flags) {
  int nt = (N + 15) / 16;
  dim3 grid((nt + 3) / 4, (M + 15) / 16);
  dim3 blk(32);
  switch (flags) {
    case 0:
      k_gemm16<0><<<grid, blk, 0, s>>>(A, B, bias, C, N, K, lda, ldb, ldc); break;
    case GF_RELU:
      k_gemm16<GF_RELU><<<grid, blk, 0, s>>>(A, B, bias, C, N, K, lda, ldb, ldc); break;
    case GF_ACC:
      k_gemm16<GF_ACC><<<grid, blk, 0, s>>>(A, B, bias, C, N, K, lda, ldb, ldc); break;
    case (GF_TRB | GF_NT):
      k_gemm16<GF_TRB | GF_NT><<<grid, blk, 0, s>>>(A, B, bias, C, N, K, lda, ldb, ldc); break;
    default:
      k_gemm16<GF_TRB><<<grid, blk, 0, s>>>(A, B, bias, C, N, K, lda, ldb, ldc); break;
  }
}

extern "C" void kernel_launch(void* const* d_in, const int* in_sizes, int n_in,
                              void* d_out, int out_size, void* d_ws, size_t ws_size,
                              hipStream_t stream) {
  const int*   seq        = (const int*)d_in[0];
  const int*   adj_rows   = (const int*)d_in[1];
  const int*   adj_cols   = (const int*)d_in[2];
  const int*   i2e        = (const int*)d_in[3];
  const float* adj_vals   = (const float*)d_in[4];
  const float* item_emb   = (const float*)d_in[5];
  const float* pos_emb    = (const float*)d_in[6];
  const float* ln_g       = (const float*)d_in[7];
  const float* ln_b       = (const float*)d_in[8];
  const float* blk_Wq     = (const float*)d_in[9];
  const float* blk_bq     = (const float*)d_in[10];
  const float* blk_Wk     = (const float*)d_in[11];
  const float* blk_bk     = (const float*)d_in[12];
  const float* blk_Wv     = (const float*)d_in[13];
  const float* blk_bv     = (const float*)d_in[14];
  const float* blk_Wo     = (const float*)d_in[15];
  const float* blk_bo     = (const float*)d_in[16];
  const float* blk_ln1_g  = (const float*)d_in[17];
  const float* blk_ln1_b  = (const float*)d_in[18];
  const float* blk_W1     = (const float*)d_in[19];
  const float* blk_b1     = (const float*)d_in[20];
  const float* blk_W2     = (const float*)d_in[21];
  const float* blk_b2     = (const float*)d_in[22];
  const float* blk_ln2_g  = (const float*)d_in[23];
  const float* blk_ln2_b  = (const float*)d_in[24];
  const float* user_kg    = (const float*)d_in[25];
  const float* ent_kg     = (const float*)d_in[26];
  const float* kg2e_tok_W = (const float*)d_in[27];
  const float* kg2e_tok_b = (const float*)d_in[28];
  const float* kgq_W      = (const float*)d_in[29];
  const float* kgk_W      = (const float*)d_in[30];
  const float* kg2e_itm_W = (const float*)d_in[31];
  const float* kg2e_itm_b = (const float*)d_in[32];
  const float* fuse_W     = (const float*)d_in[33];
  const float* fuse_b     = (const float*)d_in[34];
  float* out = (float*)d_out;

  const size_t BLD = (size_t)BB * LL * DD;       // 1,638,400
  float* ws = (float*)d_ws;
  size_t o = 0;
  float* Z  = ws + o; o += (size_t)NUM_ENT * DD; // z_entity
  float* X1 = ws + o; o += (size_t)N_NODES * DD; // GCN state 1 (region reused later)
  float* X2 = ws + o; o += (size_t)N_NODES * DD; // GCN state 2 (region reused later)
  float* MASK = ws + o; o += (size_t)BB * HH * LL * LL;
  float* ATT  = ws + o; o += (size_t)BB * HH * LL * LL;
  float* CTX  = ws + o; o += BLD;
  float* TMP  = ws + o; o += BLD;
  float* FF1  = ws + o; o += BLD * 4;
  int*   VALID = (int*)(ws + o); o += (size_t)BB * LL;
  float* UV   = ws + o; o += (size_t)BB * DD;
  float* AIV  = ws + o; o += (size_t)NUM_ITEMS * DD;
  // X1 region reuse (after z_entity finalized): token-side buffers, then item-side
  float* ENTK = X1;
  float* ENTE = X1 + BLD;
  float* QB   = X1 + 2 * BLD;
  float* KB   = X1 + 3 * BLD;
  float* AEK  = X1;                               // after MASK is built
  float* AEE  = X1 + (size_t)NUM_ITEMS * DD;
  // X2 region reuse: h / q / k / v
  float* Hb = X2;
  float* Qb = X2 + BLD;
  float* Kb = X2 + 2 * BLD;
  float* Vb = X2 + 3 * BLD;

  // ---- GCN: x1 = A x0, x2 = A x1, z = mean(x0,x1,x2)[entities] ----
  hipMemsetAsync(X1, 0, (size_t)N_NODES * DD * sizeof(float), stream);
  k_spmv<<<(NNZ + 7) / 8, 256, 0, stream>>>(adj_rows, adj_cols, adj_vals,
                                            user_kg, ent_kg, NUM_USERS, X1, NNZ);
  hipMemsetAsync(X2, 0, (size_t)N_NODES * DD * sizeof(float), stream);
  k_spmv<<<(NNZ + 7) / 8, 256, 0, stream>>>(adj_rows, adj_cols, adj_vals,
                                            X1, X1, N_NODES, X2, NNZ);
  {
    long n = (long)NUM_ENT * DD;
    k_mean<<<(unsigned)((n + 255) / 256), 256, 0, stream>>>(ent_kg, X1, X2, Z);
  }

  // ---- KG attention bias ----
  k_gather_tok<<<(BB * LL + 7) / 8, 256, 0, stream>>>(seq, i2e, Z, ENTK, VALID);
  gemm(stream, ENTK, kg2e_tok_W, kg2e_tok_b, ENTE, BB * LL, DD, DD, DD, DD, DD, 0);
  gemm(stream, ENTE, kgq_W, nullptr, QB, BB * LL, DD, DD, DD, DD, DD, 0);
  gemm(stream, ENTE, kgk_W, nullptr, KB, BB * LL, DD, DD, DD, DD, DD, 0);
  {
    long n = (long)BB * HH * LL * LL;
    k_mask<<<(unsigned)((n + 255) / 256), 256, 0, stream>>>(QB, KB, VALID, MASK);
  }

  // ---- transformer ----
  k_hinit<<<(BB * LL + 7) / 8, 256, 0, stream>>>(seq, item_emb, pos_emb, ln_g, ln_b, Hb);
  for (int i = 0; i < 2; i++) {
    const float* Wq = blk_Wq + (size_t)i * DD * DD;
    const float* Wk = blk_Wk + (size_t)i * DD * DD;
    const float* Wv = blk_Wv + (size_t)i * DD * DD;
    const float* Wo = blk_Wo + (size_t)i * DD * DD;
    const float* W1 = blk_W1 + (size_t)i * DD * 4 * DD;
    const float* W2 = blk_W2 + (size_t)i * 4 * DD * DD;
    gemm(stream, Hb, Wq, blk_bq + i * DD, Qb, BB * LL, DD, DD, DD, DD, DD, 0);
    gemm(stream, Hb, Wk, blk_bk + i * DD, Kb, BB * LL, DD, DD, DD, DD, DD, 0);
    gemm(stream, Hb, Wv, blk_bv + i * DD, Vb, BB * LL, DD, DD, DD, DD, DD, 0);
    k_attn<<<(BB * HH * LL + 3) / 4, 128, 0, stream>>>(Qb, Kb, MASK, seq, ATT);
    {
      long n = (long)BB * LL * DD;
      k_ctx<<<(unsigned)((n + 255) / 256), 256, 0, stream>>>(ATT, Vb, CTX);
    }
    gemm(stream, CTX, Wo, blk_bo + i * DD, TMP, BB * LL, DD, DD, DD, DD, DD, 0);
    k_addln<<<(BB * LL + 7) / 8, 256, 0, stream>>>(Hb, TMP, blk_ln1_g + i * DD,
                                                   blk_ln1_b + i * DD, seq, 0);
    gemm(stream, Hb, W1, blk_b1 + i * 4 * DD, FF1, BB * LL, 4 * DD, DD, DD, 4 * DD,
         4 * DD, GF_RELU);
    gemm(stream, FF1, W2, blk_b2 + i * DD, TMP, BB * LL, DD, 4 * DD, 4 * DD, DD, DD, 0);
    k_addln<<<(BB * LL + 7) / 8, 256, 0, stream>>>(Hb, TMP, blk_ln2_g + i * DD,
                                                   blk_ln2_b + i * DD, seq, 1);
  }
  k_lastpos<<<(BB + 7) / 8, 256, 0, stream>>>(seq, Hb, UV);

  // ---- item-side fusion ----
  k_gather_items<<<(NUM_ITEMS + 7) / 8, 256, 0, stream>>>(i2e, Z, AEK);
  gemm(stream, AEK, kg2e_itm_W, kg2e_itm_b, AEE, NUM_ITEMS, DD, DD, DD, DD, DD, 0);
  // all_item_vec = item_emb @ fuse_W[:64] + fuse_b  +  all_ent_e @ fuse_W[64:]
  gemm(stream, item_emb, fuse_W, fuse_b, AIV, NUM_ITEMS, DD, DD, DD, DD, DD, 0);
  gemm(stream, AEE, fuse_W + (size_t)DD * DD, nullptr, AIV, NUM_ITEMS, DD, DD, DD, DD,
       DD, GF_ACC);

  // ---- logits = user_vec @ all_item_vec^T (streaming NT stores) ----
  gemm(stream, UV, AIV, nullptr, out, BB, NUM_ITEMS, DD, DD, DD, NUM_ITEMS,
       GF_TRB | GF_NT);
}